// OSQP_37125697306999
// MI455X (gfx1250) — compile-verified
//
#include <hip/hip_runtime.h>

// Problem constants (match reference)
constexpr int BATCH = 256;
constexpr int NDIM  = 128;   // N
constexpr int MDIM  = 192;   // M
constexpr float RHO    = 0.1f;
constexpr float SIGMA_ = 1e-6f;
constexpr float ALPHA_ = 1.6f;
constexpr int   NITERS = 400;

typedef __attribute__((ext_vector_type(2))) float v2f;
typedef __attribute__((ext_vector_type(8))) float v8f;

// ---- dynamic LDS layout (offsets in floats) ----
// A (192x128) | S/Sinv (128x128) | q | l | u | x | z | y | r2 | t | xt | fcol | dslot
#define OFF_A   0
#define OFF_S   (OFF_A  + MDIM * NDIM)      // 24576
#define OFF_Q   (OFF_S  + NDIM * NDIM)      // 40960
#define OFF_L   (OFF_Q  + NDIM)
#define OFF_U   (OFF_L  + MDIM)
#define OFF_X   (OFF_U  + MDIM)
#define OFF_Z   (OFF_X  + NDIM)
#define OFF_Y   (OFF_Z  + MDIM)
#define OFF_R2  (OFF_Y  + MDIM)
#define OFF_T   (OFF_R2 + MDIM)
#define OFF_XT  (OFF_T  + NDIM)
#define OFF_FC  (OFF_XT + NDIM)
#define OFF_D   (OFF_FC + NDIM)
#define LDS_FLOATS (OFF_D + 4)              // ~170.3 KB total, < 320 KB/WGP

__global__ __launch_bounds__(256) void osqp_admm_persistent(
    const float* __restrict__ Pg_all, const float* __restrict__ qg,
    const float* __restrict__ Ag_all, const float* __restrict__ lg,
    const float* __restrict__ ug, float* __restrict__ out)
{
    extern __shared__ float smem[];
    const int b    = blockIdx.x;
    const int tid  = threadIdx.x;
    const int lane = tid & 31;
    const int wave = tid >> 5;           // 8 waves of 32 (wave32)
    const float invRho = 1.0f / RHO;

    const float* Ag = Ag_all + (size_t)b * (MDIM * NDIM);
    const float* Pg = Pg_all + (size_t)b * (NDIM * NDIM);

    float* Alds = smem + OFF_A;
    float* S    = smem + OFF_S;
    float* Qs   = smem + OFF_Q;
    float* Ls   = smem + OFF_L;
    float* Us   = smem + OFF_U;
    float* Xs   = smem + OFF_X;
    float* Zs   = smem + OFF_Z;
    float* Ys   = smem + OFF_Y;
    float* R2   = smem + OFF_R2;
    float* Tb   = smem + OFF_T;
    float* XT   = smem + OFF_XT;
    float* Fc   = smem + OFF_FC;

    // ---------------- Phase 0: stage A, q, l, u into LDS; zero state ----------------
    {
        const float4* src = (const float4*)Ag;
        float4* dst = (float4*)Alds;
        #pragma unroll 1
        for (int i = tid; i < (MDIM * NDIM) / 4; i += 256) dst[i] = src[i];
    }
    for (int i = tid; i < NDIM; i += 256) { Qs[i] = qg[b * NDIM + i]; Xs[i] = 0.0f; }
    for (int i = tid; i < MDIM; i += 256) {
        Ls[i] = lg[b * MDIM + i]; Us[i] = ug[b * MDIM + i];
        Zs[i] = 0.0f; Ys[i] = 0.0f;
    }
    __syncthreads();

    // ---------------- Phase 1: S = P + sigma*I + rho*A^T A  (WMMA fp32 16x16x4) -----
    {
        const int ti = wave;            // each wave owns one 16-row tile band of S
        const int hi = lane >> 4;       // lanes 0-15: K={0,1}; lanes 16-31: K={2,3}
        const int c  = lane & 15;
        v8f acc[8] = {};                // 8 accumulator tiles (tile-row ti, tj=0..7)
        #pragma unroll 1
        for (int k = 0; k < MDIM; k += 4) {
            v2f af;                     // A^T fragment: rows = columns of A
            af.x = Alds[(k + 2 * hi + 0) * NDIM + ti * 16 + c];
            af.y = Alds[(k + 2 * hi + 1) * NDIM + ti * 16 + c];
            #pragma unroll
            for (int tj = 0; tj < 8; ++tj) {
                v2f bf;
                bf.x = Alds[(k + 2 * hi + 0) * NDIM + tj * 16 + c];
                bf.y = Alds[(k + 2 * hi + 1) * NDIM + tj * 16 + c];
                acc[tj] = __builtin_amdgcn_wmma_f32_16x16x4_f32(
                    false, af, false, bf, (short)0, acc[tj], false, false);
            }
        }
        // C/D layout: VGPR r -> (row = r + 8*hi, col = lane&15) within the tile
        #pragma unroll
        for (int tj = 0; tj < 8; ++tj) {
            #pragma unroll
            for (int r = 0; r < 8; ++r) {
                int row = ti * 16 + r + 8 * hi;
                int col = tj * 16 + c;
                float v = RHO * acc[tj][r] + Pg[row * NDIM + col]
                        + ((row == col) ? SIGMA_ : 0.0f);
                S[row * NDIM + col] = v;
            }
        }
    }
    __syncthreads();

    // ---------------- Phase 2: in-place Gauss-Jordan inverse of S (SPD, no pivot) ---
    #pragma unroll 1
    for (int p = 0; p < NDIM; ++p) {
        if (tid == 0) smem[OFF_D] = 1.0f / S[p * NDIM + p];
        __syncthreads();
        const float d = smem[OFF_D];
        if (tid < NDIM) {
            Fc[tid] = (tid == p) ? 0.0f : S[tid * NDIM + p];      // snapshot pivot col
            S[p * NDIM + tid] = (tid == p) ? d : S[p * NDIM + tid] * d;  // scale pivot row
        }
        __syncthreads();
        {
            const int i  = tid >> 1;
            const int j0 = (tid & 1) * 64;
            if (i != p) {
                const float f = Fc[i];
                float* Srow = S + i * NDIM;
                const float* Prow = S + p * NDIM;
                #pragma unroll 4
                for (int j = j0; j < j0 + 64; ++j)
                    Srow[j] = (j == p) ? (-f * d) : (Srow[j] - f * Prow[j]);
            }
        }
        __syncthreads();
    }
    // S now holds S^{-1} (symmetric), entirely in LDS.

    // ---------------- Phase 3: 400 ADMM iterations, all state in LDS ----------------
    #pragma unroll 1
    for (int it = 0; it < NITERS; ++it) {
        if (tid < MDIM) R2[tid] = Zs[tid] - Ys[tid] * invRho;
        __syncthreads();
        if (tid < NDIM) {                       // t = sigma*x - q + rho * A^T r2
            float acc = 0.0f;
            #pragma unroll 4
            for (int m = 0; m < MDIM; ++m) acc += Alds[m * NDIM + tid] * R2[m];
            Tb[tid] = SIGMA_ * Xs[tid] - Qs[tid] + RHO * acc;
        }
        __syncthreads();
        if (tid < NDIM) {                       // xt = Sinv * t (Sinv symmetric ->
            float acc = 0.0f;                   //  read rows contiguously across lanes)
            #pragma unroll 4
            for (int k = 0; k < NDIM; ++k) acc += S[k * NDIM + tid] * Tb[k];
            XT[tid] = acc;
        }
        __syncthreads();
        if (tid < MDIM) {                       // nu, z, y updates (skewed A access)
            float acc = 0.0f;
            #pragma unroll 4
            for (int n = 0; n < NDIM; ++n) {
                int n2 = (n + tid) & (NDIM - 1);
                acc += Alds[tid * NDIM + n2] * XT[n2];
            }
            float nu  = RHO * (acc - R2[tid]);
            float zt  = Zs[tid] + (nu - Ys[tid]) * invRho;
            float zc  = ALPHA_ * zt + (1.0f - ALPHA_) * Zs[tid];
            float zcy = zc + Ys[tid] * invRho;
            float zn  = fminf(fmaxf(zcy, Ls[tid]), Us[tid]);
            Ys[tid] = Ys[tid] + RHO * (zc - zn);
            Zs[tid] = zn;
        }
        if (tid < NDIM) Xs[tid] = ALPHA_ * XT[tid] + (1.0f - ALPHA_) * Xs[tid];
        __syncthreads();
    }

    // ---------------- Phase 4: emit x ------------------------------------------------
    if (tid < NDIM) out[b * NDIM + tid] = Xs[tid];
}

extern "C" void kernel_launch(void* const* d_in, const int* in_sizes, int n_in,
                              void* d_out, int out_size, void* d_ws, size_t ws_size,
                              hipStream_t stream) {
    const float* P = (const float*)d_in[0];
    const float* q = (const float*)d_in[1];
    const float* A = (const float*)d_in[2];
    const float* l = (const float*)d_in[3];
    const float* u = (const float*)d_in[4];
    float* out = (float*)d_out;

    const size_t shmem = (size_t)LDS_FLOATS * sizeof(float);   // ~170 KB dynamic LDS
    hipFuncSetAttribute((const void*)osqp_admm_persistent,
                        hipFuncAttributeMaxDynamicSharedMemorySize, (int)shmem);
    osqp_admm_persistent<<<BATCH, 256, shmem, stream>>>(P, q, A, l, u, out);
}